// GCN_27977416966475
// MI455X (gfx1250) — compile-verified
//
#include <hip/hip_runtime.h>

typedef __attribute__((ext_vector_type(2))) float v2f;
typedef __attribute__((ext_vector_type(8))) float v8f;

#define HDIM   64
#define NGRAPH 128

// ---------------------------------------------------------------------------
// Generic fp32 WMMA GEMM: C[M,N] = A[M,K] @ B[K,N] (+bias) (optional ReLU).
// One wave32 per 16x16 tile, V_WMMA_F32_16X16X4_F32 over K in steps of 4.
// M,N multiples of 16; K multiple of 4.
// ---------------------------------------------------------------------------
__global__ void wmma_gemm_f32(const float* __restrict__ A,
                              const float* __restrict__ B,
                              const float* __restrict__ bias,
                              float* __restrict__ C,
                              int M, int N, int K, int do_relu)
{
    const int lane  = threadIdx.x & 31;
    const int warp  = threadIdx.x >> 5;
    const int tilesN   = N >> 4;
    const int numTiles = (M >> 4) * tilesN;
    const int tile = blockIdx.x * (blockDim.x >> 5) + warp;
    if (tile >= numTiles) return;            // wave-uniform: EXEC stays all-1s

    const int tm  = (tile / tilesN) << 4;    // tile row base
    const int tn  = (tile % tilesN) << 4;    // tile col base
    const int grp = lane >> 4;               // 0: K0/K1 & rows 0-7, 1: K2/K3 & rows 8-15
    const int col = lane & 15;

    v8f acc = {};
    for (int kb = 0; kb < K; kb += 4) {
        const int k0 = kb + (grp << 1);
        v2f a, b;
        a[0] = A[(tm + col) * K + k0];       // A[M=col][k0]
        a[1] = A[(tm + col) * K + k0 + 1];   // A[M=col][k0+1]
        b[0] = B[(k0    ) * N + (tn + col)]; // B[k0][N=col]
        b[1] = B[(k0 + 1) * N + (tn + col)]; // B[k0+1][N=col]
        acc = __builtin_amdgcn_wmma_f32_16x16x4_f32(
                  false, a, false, b, (short)0, acc, false, false);
    }

    const float bv = bias ? bias[tn + col] : 0.0f;
#pragma unroll
    for (int r = 0; r < 8; ++r) {
        float v = acc[r] + bv;
        if (do_relu) v = v > 0.0f ? v : 0.0f;
        C[(tm + (grp << 3) + r) * N + (tn + col)] = v;
    }
}

// ---------------------------------------------------------------------------
// Degree / normalization
// ---------------------------------------------------------------------------
__global__ void deg_init(float* __restrict__ deg, int n)
{
    int i = blockIdx.x * blockDim.x + threadIdx.x;
    if (i < n) deg[i] = 1.0f;                // +1 self loop
}

__global__ void deg_accum(const long long* __restrict__ ei, float* __restrict__ deg, int E)
{
    int e = blockIdx.x * blockDim.x + threadIdx.x;
    if (e < E) atomicAdd(&deg[(int)ei[E + e]], 1.0f);   // dst row
}

__global__ void dinv_kernel(const float* __restrict__ deg, float* __restrict__ dinv, int n)
{
    int i = blockIdx.x * blockDim.x + threadIdx.x;
    if (i < n) dinv[i] = rsqrtf(deg[i]);
}

// ---------------------------------------------------------------------------
// Edge scatter-add: agg[dst,f] += h[src,f] * dinv[src]*dinv[dst]
// 64 consecutive lanes per edge -> coalesced 256B gather of h[src,:].
// Non-returning fp32 atomics (STOREcnt path). L2-resident working set.
// ---------------------------------------------------------------------------
__global__ void edge_aggregate(const long long* __restrict__ ei,
                               const float* __restrict__ h,
                               const float* __restrict__ dinv,
                               float* __restrict__ agg, int E)
{
    int idx = blockIdx.x * blockDim.x + threadIdx.x;   // E*64 <= 51.2M, fits int
    if (idx >= E * HDIM) return;
    const int e = idx >> 6;
    const int f = idx & (HDIM - 1);
    const int s = (int)ei[e];
    const int d = (int)ei[E + e];
    const float norm = dinv[s] * dinv[d];
    atomicAdd(&agg[d * HDIM + f], h[s * HDIM + f] * norm);
}

// out[i,f] = agg[i,f] + hpre[i,f]*dinv[i]^2 + bias[f]   (optional ReLU)
__global__ void gcn_finalize(const float* __restrict__ hpre,
                             const float* __restrict__ agg,
                             const float* __restrict__ dinv,
                             const float* __restrict__ bias,
                             float* __restrict__ out, int N, int do_relu)
{
    int idx = blockIdx.x * blockDim.x + threadIdx.x;
    if (idx >= N * HDIM) return;
    const int i = idx >> 6;
    const int f = idx & (HDIM - 1);
    const float di = dinv[i];
    float v = agg[idx] + hpre[idx] * di * di + bias[f];
    if (do_relu) v = v > 0.0f ? v : 0.0f;
    out[idx] = v;
}

// ---------------------------------------------------------------------------
// Global mean pool
// ---------------------------------------------------------------------------
__global__ void pool_accum(const float* __restrict__ h,
                           const long long* __restrict__ batch,
                           float* __restrict__ sums, float* __restrict__ cnts, int N)
{
    int idx = blockIdx.x * blockDim.x + threadIdx.x;
    if (idx >= N * HDIM) return;
    const int i = idx >> 6;
    const int f = idx & (HDIM - 1);
    const int g = (int)batch[i];
    atomicAdd(&sums[g * HDIM + f], h[idx]);
    if (f == 0) atomicAdd(&cnts[g], 1.0f);
}

__global__ void pool_divide(const float* __restrict__ sums,
                            const float* __restrict__ cnts,
                            float* __restrict__ reps)
{
    int idx = blockIdx.x * blockDim.x + threadIdx.x;
    if (idx >= NGRAPH * HDIM) return;
    const int g = idx >> 6;
    reps[idx] = sums[idx] / fmaxf(cnts[g], 1.0f);
}

// ---------------------------------------------------------------------------
extern "C" void kernel_launch(void* const* d_in, const int* in_sizes, int n_in,
                              void* d_out, int out_size, void* d_ws, size_t ws_size,
                              hipStream_t stream)
{
    const float*     x     = (const float*)d_in[0];
    const long long* ei    = (const long long*)d_in[1];   // int64 [2,E]
    const long long* batch = (const long long*)d_in[2];   // int64 [N]
    const float*     W1    = (const float*)d_in[3];
    const float*     b1    = (const float*)d_in[4];
    const float*     W2    = (const float*)d_in[5];
    const float*     b2    = (const float*)d_in[6];
    const float*     Wc1   = (const float*)d_in[7];
    const float*     bc1   = (const float*)d_in[8];
    const float*     Wc2   = (const float*)d_in[9];
    const float*     bc2   = (const float*)d_in[10];

    const int N = in_sizes[0] / HDIM;   // 50000
    const int E = in_sizes[1] / 2;      // 800000
    const int C = 16;

    // d_out layout: h [N*64] | graph_reps [128*64] | logits [128*16]
    float* out_h    = (float*)d_out;
    float* out_reps = out_h + (size_t)N * HDIM;
    float* out_log  = out_reps + NGRAPH * HDIM;

    // Workspace carve-up (~26.5 MB)
    char* p = (char*)d_ws;
    float* deg  = (float*)p;  p += sizeof(float) * (size_t)N;
    float* dinv = (float*)p;  p += sizeof(float) * (size_t)N;
    float* bufA = (float*)p;  p += sizeof(float) * (size_t)N * HDIM; // h_pre
    float* bufB = (float*)p;  p += sizeof(float) * (size_t)N * HDIM; // agg / h1_relu
    float* sums = (float*)p;  p += sizeof(float) * NGRAPH * HDIM;
    float* cnts = (float*)p;  p += sizeof(float) * NGRAPH;
    float* hid  = (float*)p;  p += sizeof(float) * NGRAPH * HDIM;
    (void)ws_size; (void)n_in; (void)out_size;

    const int BT = 256;
    const int nodeElems = N * HDIM;
    const int edgeElems = E * HDIM;

    // ---- degrees & dinv (shared by both layers) ----
    deg_init  <<<(N + BT - 1) / BT, BT, 0, stream>>>(deg, N);
    deg_accum <<<(E + BT - 1) / BT, BT, 0, stream>>>(ei, deg, E);
    dinv_kernel<<<(N + BT - 1) / BT, BT, 0, stream>>>(deg, dinv, N);

    // ---- layer 1: h1 = x @ W1 ; aggregate ; +self-loop +b1 ; ReLU ----
    {
        int tiles  = (N / 16) * (HDIM / 16);
        int blocks = (tiles + 3) / 4;                       // 4 waves / 128-thr block
        wmma_gemm_f32<<<blocks, 128, 0, stream>>>(x, W1, nullptr, bufA,
                                                  N, HDIM, HDIM, 0);
    }
    hipMemsetAsync(bufB, 0, sizeof(float) * (size_t)nodeElems, stream);
    edge_aggregate<<<(edgeElems + BT - 1) / BT, BT, 0, stream>>>(ei, bufA, dinv, bufB, E);
    gcn_finalize  <<<(nodeElems + BT - 1) / BT, BT, 0, stream>>>(bufA, bufB, dinv, b1,
                                                                 bufB, N, /*relu=*/1);

    // ---- layer 2: h2 = h1_relu @ W2 ; aggregate ; +self-loop +b2 ----
    {
        int tiles  = (N / 16) * (HDIM / 16);
        int blocks = (tiles + 3) / 4;
        wmma_gemm_f32<<<blocks, 128, 0, stream>>>(bufB, W2, nullptr, bufA,
                                                  N, HDIM, HDIM, 0);
    }
    hipMemsetAsync(bufB, 0, sizeof(float) * (size_t)nodeElems, stream);
    edge_aggregate<<<(edgeElems + BT - 1) / BT, BT, 0, stream>>>(ei, bufA, dinv, bufB, E);
    gcn_finalize  <<<(nodeElems + BT - 1) / BT, BT, 0, stream>>>(bufA, bufB, dinv, b2,
                                                                 out_h, N, /*relu=*/0);

    // ---- global mean pool ----
    hipMemsetAsync(sums, 0, sizeof(float) * NGRAPH * HDIM, stream);
    hipMemsetAsync(cnts, 0, sizeof(float) * NGRAPH, stream);
    pool_accum <<<(nodeElems + BT - 1) / BT, BT, 0, stream>>>(out_h, batch, sums, cnts, N);
    pool_divide<<<(NGRAPH * HDIM + BT - 1) / BT, BT, 0, stream>>>(sums, cnts, out_reps);

    // ---- classifier: relu(reps@Wc1+bc1) @ Wc2 + bc2 ----
    {
        int tiles  = (NGRAPH / 16) * (HDIM / 16);           // 32 tiles
        int blocks = (tiles + 3) / 4;
        wmma_gemm_f32<<<blocks, 128, 0, stream>>>(out_reps, Wc1, bc1, hid,
                                                  NGRAPH, HDIM, HDIM, 1);
    }
    {
        int tiles  = (NGRAPH / 16) * (C / 16);              // 8 tiles
        int blocks = (tiles + 3) / 4;
        wmma_gemm_f32<<<blocks, 128, 0, stream>>>(hid, Wc2, bc2, out_log,
                                                  NGRAPH, C, HDIM, 0);
    }
}